// CorrelationModule_5652176961635
// MI455X (gfx1250) — compile-verified
//
#include <hip/hip_runtime.h>

// ---------------------------------------------------------------------------
// CDNA5 (gfx1250) correlation via V_WMMA_F32_16X16X4_F32
//
// corr_2d[b,y,x] = sum_{k,i,j} piece[b,k,i,j] * comp[b,k,y+i,x+j]
// corr_1d[b,x]   = sum_{k,i,j} piece[b,(k+32)&63,i,j] * strip[b,k,i,x+j]
//
// Decomposition: g[i,y',x] = sum_{k,j} piece[k,i,j]*feat[k,y',x+j]
//   (16 x 1024 x Nx GEMM -> WMMA M=i, N=x, 256 K=4 steps)
// then out[y,x] = sum_i g[i, y+i, x]   (corr_2d)
//      out1[x]  = sum_i g[i, i,   x]   (corr_1d, rolled piece)
// ---------------------------------------------------------------------------

typedef float v2f __attribute__((ext_vector_type(2)));
typedef float v8f __attribute__((ext_vector_type(8)));

#define B_        16
#define K_        64
#define HF        16      // template height
#define WF        16      // template width
#define W_        80      // feature width
#define LX        65      // output x extent
#define LY        25      // output y extent (2d)
#define FEAT_PAD  96      // zero-padded LDS row width (>= 64+15+15+1)
#define ROWS_PER_WG 4
#define WAVES     5       // one 16-wide x-tile per wave (5*16 = 80 cols)
#define TPB       (WAVES * 32)

#define PIECE_FLOATS (K_ * HF * WF)        // 16384
#define FEAT_FLOATS  (K_ * FEAT_PAD)       // 6144
#define G2D_FLOATS   (B_ * 16 * 40 * (size_t)W_)  // 819200
#define G1D_FLOATS   (B_ * 16 * 16 * (size_t)W_)  // 327680

// Phase 1: g[((b*16+i)*Hfeat + row)*80 + x] = sum_{k,j} pieceRoll[k,i,j]*feat[k,row,x+j]
__global__ __launch_bounds__(TPB)
void corr_phase1_wmma(const float* __restrict__ piece,
                      const float* __restrict__ feat,
                      float* __restrict__ g,
                      int Hfeat, int kroll)
{
    extern __shared__ float smem[];
    float* sPiece = smem;                    // [64][16][16]   = 64 KB
    float* sFeat  = smem + PIECE_FLOATS;     // [64][FEAT_PAD] = 24 KB

    const int b    = blockIdx.y;
    const int row0 = blockIdx.x * ROWS_PER_WG;
    const int tid  = threadIdx.x;
    const int wid  = tid >> 5;               // wave id: x-tile
    const int lane = tid & 31;
    const int half = lane >> 4;              // 0: kk={0,1}, 1: kk={2,3}
    const int li   = lane & 15;              // M index (A) / N index (B)
    const int x0   = wid * 16;

    // Stage the (channel-rolled) template once per workgroup.
    const float* pieceB = piece + (size_t)b * PIECE_FLOATS;
    for (int idx = tid; idx < PIECE_FLOATS; idx += TPB) {
        int k    = idx >> 8;                 // /256
        int rem  = idx & 255;                // i*16+j
        int ksrc = (k + kroll) & (K_ - 1);
        sPiece[idx] = pieceB[ksrc * 256 + rem];
    }

    // Per-lane fragment base offsets within one k-slab.
    const int aBase = li * WF + 2 * half;    // piece[k][i=li][j = jq*4 + 2*half (+1)]
    const int bBase = x0 + li + 2 * half;    // feat [k][x0+li + jq*4 + 2*half (+1)]

    for (int r = 0; r < ROWS_PER_WG; ++r) {
        const int row = row0 + r;
        const float* featRow = feat + ((size_t)(b * K_) * Hfeat + row) * W_;

        __syncthreads();   // protect sFeat reuse / first-iter piece load
        for (int idx = tid; idx < FEAT_FLOATS; idx += TPB) {
            int k = idx / FEAT_PAD;
            int x = idx - k * FEAT_PAD;
            sFeat[idx] = (x < W_) ? featRow[(size_t)k * Hfeat * W_ + x] : 0.0f;
        }
        __syncthreads();

        v8f c = {0.f, 0.f, 0.f, 0.f, 0.f, 0.f, 0.f, 0.f};
        const float* ap = sPiece + aBase;
        const float* bp = sFeat  + bBase;

        // reduction r=(k,j): 64 k-slabs x 4 chunks of K=4
        for (int k = 0; k < K_; ++k) {
            #pragma unroll
            for (int jq = 0; jq < 4; ++jq) {
                v2f a, bb;
                a.x  = ap[jq * 4];
                a.y  = ap[jq * 4 + 1];
                bb.x = bp[jq * 4];
                bb.y = bp[jq * 4 + 1];
                // D = A(16x4) * B(4x16) + C   (fp32, RNE, exec all-1s)
                c = __builtin_amdgcn_wmma_f32_16x16x4_f32(
                        false, a, false, bb, (short)0, c, false, false);
            }
            ap += HF * WF;     // next k-slab of piece (256)
            bp += FEAT_PAD;    // next k-slab of feat row (96)
        }

        // C/D layout: lanes 0-15 hold M=0..7 in c[0..7], lanes 16-31 hold M=8..15
        #pragma unroll
        for (int v = 0; v < 8; ++v) {
            int i = v + 8 * half;
            g[((size_t)(b * 16 + i) * Hfeat + row) * W_ + x0 + li] = c[v];
        }
    }
}

// Phase 2: diagonal reductions of g into the final outputs.
__global__ void corr_phase2_reduce(const float* __restrict__ g, float* __restrict__ out)
{
    int tid = blockIdx.x * blockDim.x + threadIdx.x;
    const int N2 = B_ * LY * LX;             // 26000
    const int N1 = B_ * LX;                  // 1040
    if (tid < N2) {
        int b   = tid / (LY * LX);
        int rem = tid % (LY * LX);
        int y   = rem / LX;
        int x   = rem % LX;
        float s = 0.f;
        #pragma unroll
        for (int i = 0; i < 16; ++i)
            s += g[((size_t)(b * 16 + i) * 40 + (y + i)) * W_ + x];
        out[tid] = s;
    } else if (tid < N2 + N1) {
        int t2 = tid - N2;
        int b  = t2 / LX;
        int x  = t2 % LX;
        const float* g1 = g + G2D_FLOATS;
        float s = 0.f;
        #pragma unroll
        for (int i = 0; i < 16; ++i)
            s += g1[((size_t)(b * 16 + i) * 16 + i) * W_ + x];
        out[tid] = s;
    }
}

extern "C" void kernel_launch(void* const* d_in, const int* in_sizes, int n_in,
                              void* d_out, int out_size, void* d_ws, size_t ws_size,
                              hipStream_t stream)
{
    (void)in_sizes; (void)n_in; (void)out_size; (void)ws_size;

    const float* piece = (const float*)d_in[0];  // [16,64,16,16]
    const float* comp  = (const float*)d_in[1];  // [16,64,40,80]
    const float* strip = (const float*)d_in[2];  // [16,64,16,80]
    // d_in[3] = K scalar (compile-time constant here)

    float* g   = (float*)d_ws;                   // needs ~4.6 MB
    float* out = (float*)d_out;                  // 26000 + 1040 floats

    const size_t shmem = (PIECE_FLOATS + FEAT_FLOATS) * sizeof(float); // 88 KB

    // corr_2d: 40 rows -> 10 groups of 4; corr_1d: 16 rows -> 4 groups (roll=32)
    corr_phase1_wmma<<<dim3(10, B_), TPB, shmem, stream>>>(piece, comp, g, 40, 0);
    corr_phase1_wmma<<<dim3(4, B_),  TPB, shmem, stream>>>(piece, strip, g + G2D_FLOATS, 16, 32);

    const int total = B_ * LY * LX + B_ * LX;    // 27040
    corr_phase2_reduce<<<(total + 255) / 256, 256, 0, stream>>>(g, out);
}